// ModifiedMFELoss_6236292514430
// MI455X (gfx1250) — compile-verified
//
#include <hip/hip_runtime.h>

// MI455X / gfx1250: wave32, WMMA (not MFMA).
// Memory-bound reduction: 201 MB read once -> ~8.6 us floor @ 23.3 TB/s.
// Strategy: NT b128 streaming loads, v_exp_f32 softmax, per-lane partials,
// wave-level reduction via V_WMMA_F32_16X16X4_F32 (f32-exact), one
// global_atomic_add_f32 per wave, tiny finalize kernel.

typedef __attribute__((ext_vector_type(2))) float v2f;
typedef __attribute__((ext_vector_type(4))) float v4f;
typedef __attribute__((ext_vector_type(8))) float v8f;

#define MFE_ALPHA 2.0f

__global__ void mfe_zero(float* __restrict__ acc) {
    if (threadIdx.x < 4) acc[threadIdx.x] = 0.0f;
}

// acc[0]=fne_sum, acc[1]=fpe_sum, acc[2]=n_others
__global__ __launch_bounds__(256) void mfe_partial(
    const v4f* __restrict__ outs,
    const long long* __restrict__ tgt,
    float* __restrict__ acc,
    int batch)
{
    const int tid    = blockIdx.x * blockDim.x + threadIdx.x;
    const int stride = gridDim.x * blockDim.x;

    float fne = 0.0f, fpe = 0.0f, cnt = 0.0f;

    for (int i = tid; i < batch; i += stride) {
        // Streaming, read-once data: non-temporal so we don't churn L2.
        v4f o = __builtin_nontemporal_load(&outs[i]);
        int t = (int)__builtin_nontemporal_load(&tgt[i]);

        // Numerically stable softmax over 4 classes.
        float m  = fmaxf(fmaxf(o.x, o.y), fmaxf(o.z, o.w));
        float e0 = __expf(o.x - m);
        float e1 = __expf(o.y - m);
        float e2 = __expf(o.z - m);
        float e3 = __expf(o.w - m);
        float inv = __frcp_rn(e0 + e1 + e2 + e3);
        float p0 = e0 * inv, p1 = e1 * inv, p2 = e2 * inv, p3 = e3 * inv;

        // fpe weights on (happy, sad, angry) squared probs per target branch:
        //   t==0 (others): 1,1,1 ; t==1: 0,1,1 ; t==2: 1,0,1 ; else: 1,1,0
        float wh = (t == 1) ? 0.0f : 1.0f;
        float ws = (t == 2) ? 0.0f : 1.0f;
        float wa = (t == 3) ? 0.0f : 1.0f;
        fpe += wh * (p1 * p1) + ws * (p2 * p2) + wa * (p3 * p3);

        if (t == 0) {
            float dd = p0 - 1.0f;
            fne += dd * dd;
            cnt += 1.0f;
        }
    }

    // ---- wave32 reduction via V_WMMA_F32_16X16X4_F32 -----------------------
    // A-matrix (16x4 f32) layout: lanes 0-15 hold {K=0,K=1} in VGPR0/1,
    // lanes 16-31 hold {K=2,K=3}. So a=(fne,fpe) makes columns K∈{0,2}=fne,
    // K∈{1,3}=fpe. Selector B maps fne->output col 0, fpe->output col 1:
    //   B[0][0]=B[2][0]=1 and B[1][1]=B[3][1]=1 (identical per 16-lane half).
    // D (f32 16x16): lane n<16 holds D[0..7][n] in 8 VGPRs, lane n>=16 holds
    // D[8..15][n-16]; so column c lives in lanes {c, c+16}.
    const int lane = threadIdx.x & 31;

    v2f b;
    b.x = ((lane & 15) == 0) ? 1.0f : 0.0f;
    b.y = ((lane & 15) == 1) ? 1.0f : 0.0f;
    v8f c = {};

    v2f a;
    a.x = fne;
    a.y = fpe;
    v8f d = __builtin_amdgcn_wmma_f32_16x16x4_f32(
        /*neg_a=*/false, a, /*neg_b=*/false, b,
        /*c_mod=*/(short)0, c, /*reuse_a=*/false, /*reuse_b=*/false);
    float s = d[0] + d[1] + d[2] + d[3] + d[4] + d[5] + d[6] + d[7];
    s += __shfl_down(s, 16, 32);   // lane0: wave fne total, lane1: wave fpe total

    v2f a2;
    a2.x = cnt;
    a2.y = 0.0f;
    v8f d2 = __builtin_amdgcn_wmma_f32_16x16x4_f32(
        false, a2, false, b, (short)0, c, false, false);
    float s2 = d2[0] + d2[1] + d2[2] + d2[3] + d2[4] + d2[5] + d2[6] + d2[7];
    s2 += __shfl_down(s2, 16, 32); // lane0: wave n_others total

    if (lane == 0) {
        atomicAdd(&acc[0], s);     // fne
        atomicAdd(&acc[2], s2);    // n_others
    } else if (lane == 1) {
        atomicAdd(&acc[1], s);     // fpe
    }
}

__global__ void mfe_final(const float* __restrict__ acc,
                          float* __restrict__ out, float batchf) {
    float fne = acc[0];
    float fpe = acc[1];
    float n   = acc[2];
    float fpe_num = 3.0f * n + 2.0f * (batchf - n);
    out[0] = (MFE_ALPHA / 3.0f) * (fpe / fpe_num) + MFE_ALPHA * (fne / n);
}

extern "C" void kernel_launch(void* const* d_in, const int* in_sizes, int n_in,
                              void* d_out, int out_size, void* d_ws, size_t ws_size,
                              hipStream_t stream) {
    const v4f*        outs = (const v4f*)d_in[0];        // [B,4] f32
    const long long*  tgt  = (const long long*)d_in[1];  // [B]   i64
    const int batch = in_sizes[1];
    float* acc = (float*)d_ws;                           // 3 accumulators

    mfe_zero<<<1, 32, 0, stream>>>(acc);

    const int threads = 256;                 // 8 full wave32s -> EXEC all-ones
    const int spt     = 8;                   // samples per thread
    int blocks = (batch + threads * spt - 1) / (threads * spt);
    if (blocks < 1) blocks = 1;
    mfe_partial<<<blocks, threads, 0, stream>>>(outs, tgt, acc, batch);

    mfe_final<<<1, 1, 0, stream>>>(acc, (float*)d_out, (float)batch);
}